// LengthRegulator_1013612281955
// MI455X (gfx1250) — compile-verified
//
#include <hip/hip_runtime.h>

// Problem constants (match the reference)
#define BSZ   32
#define TLEN  512
#define DDIM  256
#define PNUM  8
#define MELMX 3584          // TLEN * (PNUM-1)
#define MROWS (BSZ*TLEN)    // 16384 rows of x flattened

typedef float v2f __attribute__((ext_vector_type(2)));
typedef float v8f __attribute__((ext_vector_type(8)));

// ---------------------------------------------------------------------------
// Kernel A: fused  h = relu(x @ W1^T + b1)  ->  logits = h @ W2^T + b2
//           -> dur = argmax_p logits
// One block = 256 threads (8 waves), computes 32 rows x all 256 cols of h in
// registers via V_WMMA_F32_16X16X4_F32, stages h in LDS, then does the P=8
// dot products + shuffle argmax.  h never goes to global memory.
// ---------------------------------------------------------------------------
__global__ __launch_bounds__(256)
void dur_kernel(const float* __restrict__ x,   // (MROWS, DDIM)
                const float* __restrict__ W1,  // (DDIM, DDIM) row-major (e,d)
                const float* __restrict__ b1,  // (DDIM)
                const float* __restrict__ W2,  // (PNUM, DDIM)
                const float* __restrict__ b2,  // (PNUM)
                int* __restrict__ dur)         // (MROWS)
{
    __shared__ __align__(16) float hbuf[32 * 260];   // 32 rows, padded stride

    const int tid  = threadIdx.x;
    const int lane = tid & 31;
    const int w    = tid >> 5;          // wave id 0..7
    const int half = lane >> 4;         // 0/1 -> K sub-pair per ISA layout
    const int l16  = lane & 15;
    const int m0   = blockIdx.x * 32;   // 512 blocks cover 16384 rows

    // --- phase 1: WMMA GEMM, 2 M-subtiles x 2 N-subtiles per wave ----------
    v8f acc[2][2] = {};

    const float* arow0 = x  + (size_t)(m0 + l16) * DDIM;            // msub 0
    const float* arow1 = arow0 + 16 * DDIM;                         // msub 1
    const float* brow0 = W1 + (size_t)(w * 32 + l16) * DDIM;        // nsub 2w
    const float* brow1 = brow0 + 16 * DDIM;                         // nsub 2w+1
    const int kofs = 2 * half;

    #pragma unroll 4
    for (int k = 0; k < DDIM; k += 4) {
        v2f a0 = *(const v2f*)(arow0 + k + kofs);
        v2f a1 = *(const v2f*)(arow1 + k + kofs);
        v2f b0 = *(const v2f*)(brow0 + k + kofs);
        v2f b1v = *(const v2f*)(brow1 + k + kofs);
        acc[0][0] = __builtin_amdgcn_wmma_f32_16x16x4_f32(false, a0, false, b0,  (short)0, acc[0][0], false, false);
        acc[0][1] = __builtin_amdgcn_wmma_f32_16x16x4_f32(false, a0, false, b1v, (short)0, acc[0][1], false, false);
        acc[1][0] = __builtin_amdgcn_wmma_f32_16x16x4_f32(false, a1, false, b0,  (short)0, acc[1][0], false, false);
        acc[1][1] = __builtin_amdgcn_wmma_f32_16x16x4_f32(false, a1, false, b1v, (short)0, acc[1][1], false, false);
    }

    // bias + ReLU, scatter C/D fragments (D[v] = C[v + 8*half][l16]) to LDS
    #pragma unroll
    for (int ms = 0; ms < 2; ++ms) {
        #pragma unroll
        for (int j = 0; j < 2; ++j) {
            const int col  = w * 32 + j * 16 + l16;
            const float bv = b1[col];
            #pragma unroll
            for (int v = 0; v < 8; ++v) {
                const int row_local = ms * 16 + v + 8 * half;
                float hv = acc[ms][j][v] + bv;
                hbuf[row_local * 260 + col] = hv > 0.0f ? hv : 0.0f;
            }
        }
    }
    __syncthreads();

    // --- phase 2: logits + argmax ------------------------------------------
    const int row_local = tid >> 3;      // 0..31
    const int p         = tid & 7;       // 0..7
    const float4* hr = (const float4*)(hbuf + row_local * 260);
    const float4* wr = (const float4*)(W2 + p * DDIM);
    float accl = b2[p];
    #pragma unroll 8
    for (int e = 0; e < DDIM / 4; ++e) {
        float4 h4 = hr[e];
        float4 w4 = wr[e];
        accl += h4.x * w4.x + h4.y * w4.y + h4.z * w4.z + h4.w * w4.w;
    }

    // argmax across the 8 p-lanes (first-max tie-break, like jnp.argmax)
    float v  = accl;
    int   bi = p;
    #pragma unroll
    for (int off = 4; off >= 1; off >>= 1) {
        float ov = __shfl_xor(v,  off, 8);
        int   oi = __shfl_xor(bi, off, 8);
        if (ov > v || (ov == v && oi < bi)) { v = ov; bi = oi; }
    }
    if (p == 0) dur[m0 + row_local] = bi;
}

// ---------------------------------------------------------------------------
// Kernel B: per-batch inclusive scan of dur -> cs   (32 blocks x 512 threads)
// ---------------------------------------------------------------------------
__global__ __launch_bounds__(512)
void scan_kernel(const int* __restrict__ dur, int* __restrict__ cs)
{
    __shared__ int s[TLEN];
    const int b   = blockIdx.x;
    const int tid = threadIdx.x;
    s[tid] = dur[b * TLEN + tid];
    __syncthreads();
    #pragma unroll
    for (int off = 1; off < TLEN; off <<= 1) {
        int v = (tid >= off) ? s[tid - off] : 0;
        __syncthreads();
        s[tid] += v;
        __syncthreads();
    }
    cs[b * TLEN + tid] = s[tid];
}

// ---------------------------------------------------------------------------
// Kernel C: length-regulate gather.  grid = (MELMX/16, BSZ), 256 threads.
// Stage cs[b] in LDS, upper_bound search per mel row, then 64-lane float4
// row copies (1 KB per row) — pure streaming to HBM.
// ---------------------------------------------------------------------------
__global__ __launch_bounds__(256)
void gather_kernel(const float* __restrict__ x,   // (BSZ*TLEN, DDIM)
                   const int*   __restrict__ cs,  // (BSZ, TLEN)
                   float*       __restrict__ out) // (BSZ, MELMX, DDIM)
{
    __shared__ int cs_s[TLEN];
    __shared__ int src_s[16];

    const int tid    = threadIdx.x;
    const int b      = blockIdx.y;
    const int t_base = blockIdx.x * 16;

    cs_s[tid]       = cs[b * TLEN + tid];
    cs_s[tid + 256] = cs[b * TLEN + tid + 256];
    __syncthreads();

    if (tid < 16) {
        const int t     = t_base + tid;
        const int total = cs_s[TLEN - 1];
        int lo = 0, hi = TLEN;                 // searchsorted side='right'
        while (lo < hi) {
            int mid = (lo + hi) >> 1;
            if (cs_s[mid] <= t) lo = mid + 1; else hi = mid;
        }
        int src = lo > (TLEN - 1) ? (TLEN - 1) : lo;
        src_s[tid] = (t < total) ? src : -1;   // -1 => masked (write zeros)
    }
    __syncthreads();

    const float4* x4   = (const float4*)x;
    float4*       out4 = (float4*)out;
    const int lane64 = tid & 63;
    const int sub    = tid >> 6;               // 4 rows in flight per pass

    #pragma unroll
    for (int it = 0; it < 4; ++it) {
        const int r = it * 4 + sub;
        const int s = src_s[r];
        float4 val = make_float4(0.0f, 0.0f, 0.0f, 0.0f);
        if (s >= 0)
            val = x4[((size_t)b * TLEN + s) * (DDIM / 4) + lane64];
        out4[((size_t)b * MELMX + (t_base + r)) * (DDIM / 4) + lane64] = val;
    }
}

// ---------------------------------------------------------------------------
extern "C" void kernel_launch(void* const* d_in, const int* in_sizes, int n_in,
                              void* d_out, int out_size, void* d_ws, size_t ws_size,
                              hipStream_t stream)
{
    const float* x  = (const float*)d_in[0];   // (B,T,D)
    const float* W1 = (const float*)d_in[1];   // (D,D)
    const float* b1 = (const float*)d_in[2];   // (D)
    const float* W2 = (const float*)d_in[3];   // (P,D)
    const float* b2 = (const float*)d_in[4];   // (P)
    // d_in[5] = mel_max_length scalar (compile-time constant MELMX here)

    int* dur = (int*)d_ws;                     // 16384 ints
    int* cs  = dur + MROWS;                    // 16384 ints (128 KB total ws)

    dur_kernel  <<<dim3(MROWS / 32), 256, 0, stream>>>(x, W1, b1, W2, b2, dur);
    scan_kernel <<<dim3(BSZ),        512, 0, stream>>>(dur, cs);
    gather_kernel<<<dim3(MELMX / 16, BSZ), 256, 0, stream>>>(x, cs, (float*)d_out);
}